// CausalSelfAttention_8409545966069
// MI455X (gfx1250) — compile-verified
//
#include <hip/hip_runtime.h>
#include <math.h>

typedef __attribute__((ext_vector_type(2))) float v2f;
typedef __attribute__((ext_vector_type(4))) float v4f;
typedef __attribute__((ext_vector_type(8))) float v8f;

// D = A(16x4) * B(4x16) + C, full fp32
#define WMMA_F32(a, b, c) \
  __builtin_amdgcn_wmma_f32_16x16x4_f32(false, (a), false, (b), (short)0, (c), false, false)

static constexpr int Bb = 2, Tt = 2048, Dd = 2048, Hh = 16, DhC = 128;
static constexpr int ROWSTRIDE = 3 * Dd; // 6144 floats between consecutive t in qkv buffer

// ---------------------------------------------------------------------------
// 32x32-tile transpose: WT[N][K] = W[K][N]
// ---------------------------------------------------------------------------
__global__ __launch_bounds__(256)
void transpose_kernel(const float* __restrict__ W, float* __restrict__ WT,
                      int Krows, int Ncols)
{
  __shared__ float tile[32][33];
  const int tx = threadIdx.x, ty = threadIdx.y;      // (32, 8)
  const int n0 = blockIdx.x * 32, k0 = blockIdx.y * 32;
#pragma unroll
  for (int i = 0; i < 32; i += 8)
    tile[ty + i][tx] = W[(size_t)(k0 + ty + i) * Ncols + n0 + tx];
  __syncthreads();
#pragma unroll
  for (int i = 0; i < 32; i += 8)
    WT[(size_t)(n0 + ty + i) * Krows + k0 + tx] = tile[tx][ty + i];
}

// ---------------------------------------------------------------------------
// C[M,N] = A[M,K] @ B[K,N] + bias[N], with B supplied TRANSPOSED (BT[N][K]).
// One wave computes a 32(M) x 64(N) tile: 2 A-fragments x 4 B-fragments,
// 8 WMMAs per K-step of 4 against 6 contiguous b64 loads (immediate offsets).
// Fragment layout (ISA 7.12.2, 32-bit A 16x4): lane<16 holds K pair (k,k+1),
// lanes 16..31 hold (k+2,k+3); A lane -> row, B lane -> column (BT row).
// ---------------------------------------------------------------------------
__global__ __launch_bounds__(32)
void gemm_bias_32x64(const float* __restrict__ A, const float* __restrict__ BT,
                     const float* __restrict__ bias, float* __restrict__ C,
                     int M, int N, int K)
{
  const int lane = threadIdx.x;
  const int hi   = lane >> 4;
  const int lo   = lane & 15;
  const int n0   = blockIdx.x * 64;
  const int m0   = blockIdx.y * 32;

  v8f c[2][4];
  {
    float bv0 = bias[n0 + 0  + lo];
    float bv1 = bias[n0 + 16 + lo];
    float bv2 = bias[n0 + 32 + lo];
    float bv3 = bias[n0 + 48 + lo];
#pragma unroll
    for (int mi = 0; mi < 2; ++mi)
#pragma unroll
      for (int r = 0; r < 8; ++r) {
        c[mi][0][r] = bv0; c[mi][1][r] = bv1;
        c[mi][2][r] = bv2; c[mi][3][r] = bv3;
      }
  }

  const float* a0p = A  + (size_t)(m0 + lo) * K + 2 * hi;
  const float* a1p = a0p + (size_t)16 * K;
  const float* b0p = BT + (size_t)(n0 + 0  + lo) * K + 2 * hi;
  const float* b1p = BT + (size_t)(n0 + 16 + lo) * K + 2 * hi;
  const float* b2p = BT + (size_t)(n0 + 32 + lo) * K + 2 * hi;
  const float* b3p = BT + (size_t)(n0 + 48 + lo) * K + 2 * hi;

  for (int k = 0; k < K; k += 4) {
    v2f A0 = *(const v2f*)(a0p + k);
    v2f A1 = *(const v2f*)(a1p + k);
    v2f B0 = *(const v2f*)(b0p + k);
    v2f B1 = *(const v2f*)(b1p + k);
    v2f B2 = *(const v2f*)(b2p + k);
    v2f B3 = *(const v2f*)(b3p + k);
    c[0][0] = WMMA_F32(A0, B0, c[0][0]);
    c[0][1] = WMMA_F32(A0, B1, c[0][1]);
    c[0][2] = WMMA_F32(A0, B2, c[0][2]);
    c[0][3] = WMMA_F32(A0, B3, c[0][3]);
    c[1][0] = WMMA_F32(A1, B0, c[1][0]);
    c[1][1] = WMMA_F32(A1, B1, c[1][1]);
    c[1][2] = WMMA_F32(A1, B2, c[1][2]);
    c[1][3] = WMMA_F32(A1, B3, c[1][3]);
  }

#pragma unroll
  for (int mi = 0; mi < 2; ++mi) {
    float* crow = C + (size_t)(m0 + mi * 16 + 8 * hi) * N + n0 + lo;
#pragma unroll
    for (int r = 0; r < 8; ++r) {
      crow[(size_t)r * N + 0 ] = c[mi][0][r];
      crow[(size_t)r * N + 16] = c[mi][1][r];
      crow[(size_t)r * N + 32] = c[mi][2][r];
      crow[(size_t)r * N + 48] = c[mi][3][r];
    }
  }
}

// ---------------------------------------------------------------------------
// Flash attention, fp32, causal. One wave per (16-query tile, head, batch).
// Tiles staged to LDS with gfx1250 async global->LDS b128 transfers
// (ASYNCcnt-tracked), consumed after s_wait_asynccnt 0.
// qkv layout: [(b*T+t)*6144 + i3*2048 + h*128 + dh]   (i3: 0=q,1=k,2=v)
// out layout: [(b*T+t)*2048 + h*128 + dh]
// ---------------------------------------------------------------------------
#define QPAD 132   // 128 + 4 pad: conflict-free fragment reads, 16B-aligned rows
#define PPAD 18    // 16 + 2 pad: 8B aligned, distinct banks across 16 lanes

__device__ __forceinline__ void async_tile_b128(const float* gsrc, float* ldst)
{
  unsigned lds = (unsigned)(size_t)ldst;   // low 32 bits of generic addr = LDS offset
  asm volatile("global_load_async_to_lds_b128 %0, %1, off"
               :: "v"(lds), "v"(gsrc) : "memory");
}

__global__ __launch_bounds__(32)
void flash_attn(const float* __restrict__ qkv, float* __restrict__ out)
{
  __shared__ float sQ[16 * QPAD];
  __shared__ float sK[16 * QPAD];
  __shared__ float sV[16 * QPAD];
  __shared__ float sP[16 * PPAD];

  const int lane = threadIdx.x;
  const int hi = lane >> 4, lo = lane & 15;
  const int qi = blockIdx.x;          // query tile
  const int h  = blockIdx.y;
  const int b  = blockIdx.z;
  const int q0 = qi * 16;

  const float* base  = qkv + (size_t)b * Tt * ROWSTRIDE + (size_t)h * DhC;
  const float* kbase = base + Dd;       // +2048
  const float* vbase = base + 2 * Dd;   // +4096

  // stage Q tile (16 x 128) into LDS via async b128 (16B per lane per unit)
  for (int u = lane; u < 512; u += 32) {
    int row = u >> 5; int c4 = (u & 31) * 4;
    async_tile_b128(base + (size_t)(q0 + row) * ROWSTRIDE + c4, &sQ[row * QPAD + c4]);
  }

  float m_r[8], l_r[8];
#pragma unroll
  for (int r = 0; r < 8; ++r) { m_r[r] = -1e30f; l_r[r] = 0.0f; }
  v8f o[8] = {};

  const float scale = 0.08838834764831845f;  // 1/sqrt(128)

  for (int j = 0; j <= qi; ++j) {
    const int j0 = j * 16;
    __syncthreads();                          // previous tile fully consumed
    for (int u = lane; u < 512; u += 32) {    // stage K,V tiles (16 x 128 each)
      int row = u >> 5; int c4 = (u & 31) * 4;
      async_tile_b128(kbase + (size_t)(j0 + row) * ROWSTRIDE + c4, &sK[row * QPAD + c4]);
      async_tile_b128(vbase + (size_t)(j0 + row) * ROWSTRIDE + c4, &sV[row * QPAD + c4]);
    }
    asm volatile("s_wait_asynccnt 0x0" ::: "memory");
    __syncthreads();

    // S = Q @ K^T  (16x16, K-dim = Dh = 128)
    v8f s = {};
#pragma unroll
    for (int kk = 0; kk < DhC; kk += 4) {
      v2f a  = *(const v2f*)(&sQ[lo * QPAD + kk + 2 * hi]);
      v2f bf = *(const v2f*)(&sK[lo * QPAD + kk + 2 * hi]);  // B col = key row of K
      s = WMMA_F32(a, bf, s);
    }

    // online softmax update (stats replicated across each 16-lane half)
#pragma unroll
    for (int r = 0; r < 8; ++r) {
      float sv = s[r] * scale;
      int qg = q0 + r + 8 * hi;     // query row of this element
      int kg = j0 + lo;             // key column of this element
      if (kg > qg) sv = -1e30f;     // causal mask (only bites on diagonal tile)
      float rmax = sv;
      rmax = fmaxf(rmax, __shfl_xor(rmax, 1));
      rmax = fmaxf(rmax, __shfl_xor(rmax, 2));
      rmax = fmaxf(rmax, __shfl_xor(rmax, 4));
      rmax = fmaxf(rmax, __shfl_xor(rmax, 8));
      float mnew = fmaxf(m_r[r], rmax);
      float p = __expf(sv - mnew);
      float rsum = p;
      rsum += __shfl_xor(rsum, 1);
      rsum += __shfl_xor(rsum, 2);
      rsum += __shfl_xor(rsum, 4);
      rsum += __shfl_xor(rsum, 8);
      float alpha = __expf(m_r[r] - mnew);
      l_r[r] = l_r[r] * alpha + rsum;
      m_r[r] = mnew;
#pragma unroll
      for (int t = 0; t < 8; ++t) o[t][r] *= alpha;
      sP[(r + 8 * hi) * PPAD + lo] = p;   // C-layout -> row-major P in LDS
    }
    __syncthreads();

    // O += P @ V   (M=16 queries, K=16 keys, N=128 = 8 tiles of 16)
#pragma unroll
    for (int t = 0; t < 8; ++t) {
#pragma unroll
      for (int kk = 0; kk < 16; kk += 4) {
        v2f a = *(const v2f*)(&sP[lo * PPAD + kk + 2 * hi]);
        v2f bf;
        bf[0] = sV[(kk + 2 * hi + 0) * QPAD + t * 16 + lo];
        bf[1] = sV[(kk + 2 * hi + 1) * QPAD + t * 16 + lo];
        o[t] = WMMA_F32(a, bf, o[t]);
      }
    }
  }

  // normalize and write: out[(b*T + q0 + M)*D + h*128 + t*16 + lo]
  float* obase = out + (size_t)(b * Tt + q0 + 8 * hi) * Dd + (size_t)h * DhC + lo;
#pragma unroll
  for (int r = 0; r < 8; ++r) {
    float inv = 1.0f / l_r[r];
#pragma unroll
    for (int t = 0; t < 8; ++t)
      obase[(size_t)r * Dd + t * 16] = o[t][r] * inv;
  }
}

// ---------------------------------------------------------------------------
extern "C" void kernel_launch(void* const* d_in, const int* in_sizes, int n_in,
                              void* d_out, int out_size, void* d_ws, size_t ws_size,
                              hipStream_t stream)
{
  (void)in_sizes; (void)n_in; (void)out_size; (void)ws_size;
  const float* x    = (const float*)d_in[0];
  const float* Wqkv = (const float*)d_in[1];
  const float* bqkv = (const float*)d_in[2];
  const float* Wout = (const float*)d_in[3];
  const float* bout = (const float*)d_in[4];
  float* outp = (float*)d_out;

  float* qkv     = (float*)d_ws;                         // B*T*3D  (~100 MB)
  float* attnout = qkv + (size_t)Bb * Tt * 3 * Dd;       // B*T*D   (~34 MB)
  float* WqkvT   = attnout + (size_t)Bb * Tt * Dd;       // 3D x D  (~50 MB)
  float* WoutT   = WqkvT + (size_t)3 * Dd * Dd;          // D x D   (~17 MB)

  {   // transpose weights once (re-used 128-256x by the GEMMs)
    dim3 g1((3 * Dd) / 32, Dd / 32);
    transpose_kernel<<<g1, dim3(32, 8), 0, stream>>>(Wqkv, WqkvT, Dd, 3 * Dd);
    dim3 g2(Dd / 32, Dd / 32);
    transpose_kernel<<<g2, dim3(32, 8), 0, stream>>>(Wout, WoutT, Dd, Dd);
  }
  {   // QKV projection: [4096,2048] @ [2048,6144] + bqkv
    dim3 grid((3 * Dd) / 64, (Bb * Tt) / 32);
    gemm_bias_32x64<<<grid, 32, 0, stream>>>(x, WqkvT, bqkv, qkv,
                                             Bb * Tt, 3 * Dd, Dd);
  }
  {   // causal flash attention per (q-tile, head, batch)
    dim3 grid(Tt / 16, Hh, Bb);
    flash_attn<<<grid, 32, 0, stream>>>(qkv, attnout);
  }
  {   // output projection: [4096,2048] @ [2048,2048] + bout
    dim3 grid(Dd / 64, (Bb * Tt) / 32);
    gemm_bias_32x64<<<grid, 32, 0, stream>>>(attnout, WoutT, bout, outp,
                                             Bb * Tt, Dd, Dd);
  }
}